// GCNLayer_13305808683526
// MI455X (gfx1250) — compile-verified
//
#include <hip/hip_runtime.h>
#include <hip/hip_bf16.h>

// GCN layer for MI455X (gfx1250, wave32):
//   agg = segment_sum(x[row] -> col);  deg = clamp(indeg,1)
//   out = (agg/deg) @ W^T + b
// Phase 1: atomic scatter (bandwidth bound, resolves in 192MB L2)
// Phase 2: WMMA f32 16x16x4 GEMM with LDS-staged, bank-conflict-free tiles.

typedef __attribute__((ext_vector_type(2))) float v2f;
typedef __attribute__((ext_vector_type(8))) float v8f;

// ---------------------------------------------------------------- zero ws
__global__ void gcn_zero(float* __restrict__ p, int n) {
    int i = blockIdx.x * blockDim.x + threadIdx.x;
    if (i < n) p[i] = 0.0f;
}

// ---------------------------------------------------------------- scatter
// one wave (32 lanes) per edge; lane l handles features 4l..4l+3
__global__ __launch_bounds__(256) void gcn_scatter(
    const float* __restrict__ x,       // [n_nodes,128]
    const int*   __restrict__ ei,      // [2, n_edges]
    float* __restrict__ agg,           // [n_nodes,128]
    float* __restrict__ deg,           // [n_nodes]
    int n_edges)
{
    const int wid  = (blockIdx.x * blockDim.x + threadIdx.x) >> 5;
    const int lane = threadIdx.x & 31;
    if (wid >= n_edges) return;

    const int row = ei[wid];            // source node
    const int col = ei[n_edges + wid];  // destination node

    const float4 v = reinterpret_cast<const float4*>(x)[(size_t)row * 32 + lane];
    float* dst = agg + (size_t)col * 128 + lane * 4;
    atomicAdd(dst + 0, v.x);
    atomicAdd(dst + 1, v.y);
    atomicAdd(dst + 2, v.z);
    atomicAdd(dst + 3, v.w);
    if (lane == 0) atomicAdd(deg + col, 1.0f);
}

// ---------------------------------------------------------------- GEMM
// block = 256 thr = 8 waves; block tile = 128 rows x 128 cols, K = 128.
// A (=agg/deg) and W staged in LDS with padded stride 132 floats so the
// row-strided WMMA fragment fetches hit distinct banks (132 % 64 == 4).
__global__ __launch_bounds__(256) void gcn_gemm_wmma(
    const float* __restrict__ agg,     // [n_nodes,128]
    const float* __restrict__ deg,     // [n_nodes]
    const float* __restrict__ W,       // [128,128]  (out_f x in_f)
    const float* __restrict__ bias,    // [128]
    float* __restrict__ out,           // [n_nodes,128]
    int n_nodes)
{
    constexpr int PAD = 132;           // 128 + 4 floats padding per row
    __shared__ float As[128 * PAD];    // 66 KB
    __shared__ float Ws[128 * PAD];    // 66 KB  (total 132 KB of 320 KB/WGP)

    const int tid = threadIdx.x;
    const int R0  = blockIdx.x * 128;

    // cooperative, coalesced load of A (normalized) and W into LDS
    const float4* agg4 = reinterpret_cast<const float4*>(agg);
    const float4* W4   = reinterpret_cast<const float4*>(W);
#pragma unroll
    for (int i = 0; i < 16; ++i) {
        const int f4 = tid + i * 256;   // 0..4095
        const int r  = f4 >> 5;         // row 0..127
        const int c4 = f4 & 31;         // float4 column
        const int gr = R0 + r;
        float4 a = make_float4(0.f, 0.f, 0.f, 0.f);
        float invd = 0.f;
        if (gr < n_nodes) {
            a = agg4[(size_t)gr * 32 + c4];
            invd = 1.0f / fmaxf(deg[gr], 1.0f);
        }
        const float4 w = W4[r * 32 + c4];
        *reinterpret_cast<float4*>(&As[r * PAD + c4 * 4]) =
            make_float4(a.x * invd, a.y * invd, a.z * invd, a.w * invd);
        *reinterpret_cast<float4*>(&Ws[r * PAD + c4 * 4]) = w;
    }
    __syncthreads();

    // per-wave 16x16 WMMA tiles, ISA 7.12.2 fragment layouts:
    //   A 16x4 f32 : lanes 0-15 -> M=lane, K={0,1}; lanes 16-31 -> K={2,3}
    //   B 4x16 f32 : lanes 0-15 -> N=lane, K={0,1}; lanes 16-31 -> K={2,3}
    //   C/D 16x16  : VGPR v, lane L -> M = v + (L<16 ? 0 : 8), N = L&15
    const int wave = tid >> 5;
    const int lane = tid & 31;
    const int lm   = lane & 15;
    const int koff = (lane < 16) ? 0 : 2;
    const int m0   = wave * 16;

    const float* arow = &As[(m0 + lm) * PAD + koff];

#pragma unroll 1
    for (int nt = 0; nt < 8; ++nt) {
        const int n0 = nt * 16;
        const float bn = bias[n0 + lm];
        const float* brow = &Ws[(n0 + lm) * PAD + koff];

        v8f acc = {0.f, 0.f, 0.f, 0.f, 0.f, 0.f, 0.f, 0.f};
#pragma unroll
        for (int k = 0; k < 128; k += 4) {
            const v2f a = *reinterpret_cast<const v2f*>(arow + k);
            const v2f b = *reinterpret_cast<const v2f*>(brow + k);
            acc = __builtin_amdgcn_wmma_f32_16x16x4_f32(
                false, a, false, b, (short)0, acc, false, false);
        }

        const int mbase = m0 + ((lane < 16) ? 0 : 8);
#pragma unroll
        for (int v = 0; v < 8; ++v) {
            const int gm = R0 + mbase + v;
            if (gm < n_nodes)
                out[(size_t)gm * 128 + n0 + lm] = acc[v] + bn;
        }
    }
}

// ---------------------------------------------------------------- launch
extern "C" void kernel_launch(void* const* d_in, const int* in_sizes, int n_in,
                              void* d_out, int out_size, void* d_ws, size_t ws_size,
                              hipStream_t stream) {
    const float* x  = (const float*)d_in[0];   // [n_nodes,128]
    const int*   ei = (const int*)d_in[1];     // [2,n_edges]
    const float* W  = (const float*)d_in[3];   // [128,128]
    const float* b  = (const float*)d_in[4];   // [128]
    float* out = (float*)d_out;

    const int n_nodes = in_sizes[0] / 128;
    const int n_edges = in_sizes[1] / 2;

    float* agg = (float*)d_ws;                       // n_nodes*128 floats
    float* deg = agg + (size_t)n_nodes * 128;        // n_nodes floats

    const int zn = n_nodes * 128 + n_nodes;
    gcn_zero<<<(zn + 1023) / 1024, 1024, 0, stream>>>(agg, zn);

    const int edge_blocks = (n_edges + 7) / 8;       // 8 waves/block, 1 edge/wave
    gcn_scatter<<<edge_blocks, 256, 0, stream>>>(x, ei, agg, deg, n_edges);

    const int gemm_blocks = (n_nodes + 127) / 128;
    gcn_gemm_wmma<<<gemm_blocks, 256, 0, stream>>>(agg, deg, W, b, out, n_nodes);
}